// EmbeddingLayer_75763223101599
// MI455X (gfx1250) — compile-verified
//
#include <hip/hip_runtime.h>
#include <math.h>

// out[t, d] = W[input[t], d] + pe(t, d)
// pe(t, d)  = (t even) ? sin(t * w_d) : cos(t * w_d),  w_d = 10000^(-2d/D)
//
// Memory-bound (≈64 MB of traffic -> ~2.8 us at 23.3 TB/s on MI455X). Plan:
//  - 1 float4 (b128) per lane, 256 lanes cover one 4 KB row (D=1024 f32)
//  - 4 tokens per block, indices fetched with ONE uniform s_load_b128
//  - guard-free main kernel (T % 4 handled by a tail kernel) -> straight-line
//    block: 4 independent b128 gathers, PE transcendentals overlap the loads
//  - non-temporal b128 stores (output never re-read) so the 192 MB L2 stays
//    full of gathered W rows (duplicate-token reuse)

typedef float v4f __attribute__((ext_vector_type(4)));

#define TPB_TOKENS 4

__device__ __forceinline__ float reduce_2pi(float x) {
    // x - 2*pi*round(x/(2*pi)), two-term compensated -> |r| <= pi
    const float INV_2PI   = 0.15915493667125702f;
    const float TWO_PI_HI = 6.2831854820251465f;      // fp32(2*pi)
    const float TWO_PI_LO = -1.7484556000744083e-07f; // 2*pi - fp32(2*pi)
    float n = rintf(x * INV_2PI);
    float r = fmaf(n, -TWO_PI_HI, x);
    r = fmaf(n, -TWO_PI_LO, r);
    return r;
}

__device__ __forceinline__ float pe_sin(float x) { return __sinf(reduce_2pi(x)); }
__device__ __forceinline__ float pe_cos(float x) { return __cosf(reduce_2pi(x)); }

// Main kernel: exactly 4 tokens per block, no bounds checks.
__global__ __launch_bounds__(256) void embed_pe_main(
    const int*   __restrict__ idx,   // [T], T % 4 == 0 region
    const float* __restrict__ W,     // [VOCAB, D]
    float*       __restrict__ out,   // [T, D]
    int D, float c)                  // c = -2*log2(10000)/D
{
    const int d  = (int)threadIdx.x * 4;        // blockDim.x == D/4
    const int t0 = (int)blockIdx.x * TPB_TOKENS; // even by construction

    // One uniform 16B read of 4 contiguous token ids -> single s_load_b128.
    const int4 tok = *(const int4*)(idx + t0);

    // Frequencies w_d, shared by all 4 rows. v_exp_f32 computes 2^x.
    const float f0 = __builtin_amdgcn_exp2f(c * (float)(d + 0));
    const float f1 = __builtin_amdgcn_exp2f(c * (float)(d + 1));
    const float f2 = __builtin_amdgcn_exp2f(c * (float)(d + 2));
    const float f3 = __builtin_amdgcn_exp2f(c * (float)(d + 3));

    // 4 independent b128 gathers issued back-to-back.
    const v4f w0 = *(const v4f*)(W + (size_t)tok.x * (size_t)D + d);
    const v4f w1 = *(const v4f*)(W + (size_t)tok.y * (size_t)D + d);
    const v4f w2 = *(const v4f*)(W + (size_t)tok.z * (size_t)D + d);
    const v4f w3 = *(const v4f*)(W + (size_t)tok.w * (size_t)D + d);

    // PE rows: t0+0 (even->sin), t0+1 (cos), t0+2 (sin), t0+3 (cos).
    const float ta = (float)(t0 + 0), tb = (float)(t0 + 1);
    const float tc = (float)(t0 + 2), td = (float)(t0 + 3);

    v4f p0, p1, p2, p3;
    p0.x = pe_sin(ta * f0); p0.y = pe_sin(ta * f1); p0.z = pe_sin(ta * f2); p0.w = pe_sin(ta * f3);
    p1.x = pe_cos(tb * f0); p1.y = pe_cos(tb * f1); p1.z = pe_cos(tb * f2); p1.w = pe_cos(tb * f3);
    p2.x = pe_sin(tc * f0); p2.y = pe_sin(tc * f1); p2.z = pe_sin(tc * f2); p2.w = pe_sin(tc * f3);
    p3.x = pe_cos(td * f0); p3.y = pe_cos(td * f1); p3.z = pe_cos(td * f2); p3.w = pe_cos(td * f3);

    float* o0 = out + (size_t)(t0 + 0) * (size_t)D + d;
    float* o1 = out + (size_t)(t0 + 1) * (size_t)D + d;
    float* o2 = out + (size_t)(t0 + 2) * (size_t)D + d;
    float* o3 = out + (size_t)(t0 + 3) * (size_t)D + d;

    __builtin_nontemporal_store(w0 + p0, (v4f*)o0);
    __builtin_nontemporal_store(w1 + p1, (v4f*)o1);
    __builtin_nontemporal_store(w2 + p2, (v4f*)o2);
    __builtin_nontemporal_store(w3 + p3, (v4f*)o3);
}

// Tail kernel: one token per block (only launched when T % 4 != 0).
__global__ __launch_bounds__(256) void embed_pe_tail(
    const int*   __restrict__ idx,
    const float* __restrict__ W,
    float*       __restrict__ out,
    int D, int t_base, float c)
{
    const int d = (int)threadIdx.x * 4;
    const int t = t_base + (int)blockIdx.x;

    const int tok = idx[t];                     // uniform -> s_load_b32
    const float f0 = __builtin_amdgcn_exp2f(c * (float)(d + 0));
    const float f1 = __builtin_amdgcn_exp2f(c * (float)(d + 1));
    const float f2 = __builtin_amdgcn_exp2f(c * (float)(d + 2));
    const float f3 = __builtin_amdgcn_exp2f(c * (float)(d + 3));

    const v4f w = *(const v4f*)(W + (size_t)tok * (size_t)D + d);
    const float tf = (float)t;

    v4f p;
    if ((t & 1) == 0) {
        p.x = pe_sin(tf * f0); p.y = pe_sin(tf * f1);
        p.z = pe_sin(tf * f2); p.w = pe_sin(tf * f3);
    } else {
        p.x = pe_cos(tf * f0); p.y = pe_cos(tf * f1);
        p.z = pe_cos(tf * f2); p.w = pe_cos(tf * f3);
    }
    __builtin_nontemporal_store(w + p, (v4f*)(out + (size_t)t * (size_t)D + d));
}

extern "C" void kernel_launch(void* const* d_in, const int* in_sizes, int n_in,
                              void* d_out, int out_size, void* d_ws, size_t ws_size,
                              hipStream_t stream) {
    (void)n_in; (void)d_ws; (void)ws_size;
    const int*   idx = (const int*)d_in[0];    // input_ids [T], int32
    const float* W   = (const float*)d_in[1];  // [VOCAB, D], f32
    float*       out = (float*)d_out;          // [T, D], f32

    const int T = in_sizes[0];                 // 8192
    const int D = out_size / T;                // 1024
    const float c = -2.0f * log2f(10000.0f) / (float)D;

    const int threads = D / 4;                 // 256: one float4 per lane
    const int main_blocks = T / TPB_TOKENS;    // 2048
    const int rem = T - main_blocks * TPB_TOKENS;

    if (main_blocks > 0) {
        embed_pe_main<<<main_blocks, threads, 0, stream>>>(idx, W, out, D, c);
    }
    if (rem > 0) {                             // size-derived -> deterministic
        embed_pe_tail<<<rem, threads, 0, stream>>>(idx, W, out, D,
                                                   main_blocks * TPB_TOKENS, c);
    }
}